// Init_50448685859053
// MI455X (gfx1250) — compile-verified
//
#include <hip/hip_runtime.h>

typedef float v2f __attribute__((ext_vector_type(2)));
typedef float v8f __attribute__((ext_vector_type(8)));

#define N_EDGES     500000
#define HIDDEN      128
#define N_TILES     (N_EDGES / 16)   // 31250 exactly
#define WAVES_PB    8
#define N_GROUPS    ((N_TILES + WAVES_PB - 1) / WAVES_PB)   // 3907
#define GRID_BLKS   1024             // persistent-style grid; grid-stride over groups
#define WLIN_STRIDE 388              // dwords; %64==4 -> conflict-free b64 B-frag reads
#define CAT_STRIDE  66               // dwords per lane; %64==2 -> conflict-free b64 A reads

__global__ __launch_bounds__(256, 1)
void edge_mlp_kernel(const int* __restrict__ xin,
                     const float* __restrict__ rbf,
                     const int* __restrict__ iidx,
                     const int* __restrict__ jidx,
                     const float* __restrict__ emb,
                     const float* __restrict__ w0,
                     const float* __restrict__ b0,
                     const float* __restrict__ wlin,
                     const float* __restrict__ blin,
                     const float* __restrict__ w1,
                     float* __restrict__ out1,
                     float* __restrict__ out2)
{
    // ---- LDS (CDNA5: 320KB per WGP; total here ~269KB, 1 block/WGP, 8 waves) ----
    __shared__ float s_wlin[HIDDEN * WLIN_STRIDE];           // 198656 B, [h][k] padded
    __shared__ float s_cat[WAVES_PB][32 * CAT_STRIDE];       //  67584 B, per-lane A slices
    __shared__ float s_w0p[HIDDEN * 8];                      // [c] = {w0 r0..5, b0, 0}
    __shared__ float s_w1p[HIDDEN * 8];                      // [c] = {w1 r0..5, 0, 0}
    __shared__ float s_blin[HIDDEN];

    const int tid = threadIdx.x;

    // ---- stage w_lin into LDS once per block via CDNA5 async LDS DMA ----
    // Each lane moves 16B: global row-major [128][384] -> LDS rows padded to 388 dw.
    {
        for (int c4 = tid; c4 < HIDDEN * 96; c4 += 256) {    // 96 dword-quads per row
            int h  = c4 / 96;
            int kq = c4 - h * 96;
            unsigned int lds_off = (unsigned int)(size_t)(&s_wlin[h * WLIN_STRIDE + kq * 4]);
            unsigned int g_off   = (unsigned int)((h * 384 + kq * 4) * 4);
            asm volatile("global_load_async_to_lds_b128 %0, %1, %2"
                         :: "v"(lds_off), "v"(g_off), "s"(wlin)
                         : "memory");
        }
        asm volatile("s_wait_asynccnt 0" ::: "memory");       // LDS writes landed
    }
    if (tid < HIDDEN) {
        int c = tid;
        #pragma unroll
        for (int r = 0; r < 6; ++r) {
            s_w0p[c * 8 + r] = w0[c * 6 + r];
            s_w1p[c * 8 + r] = w1[c * 6 + r];
        }
        s_w0p[c * 8 + 6] = b0[c];
        s_w0p[c * 8 + 7] = 0.0f;
        s_w1p[c * 8 + 6] = 0.0f;
        s_w1p[c * 8 + 7] = 0.0f;
        s_blin[c] = blin[c];
    }
    __syncthreads();

    const int wave = tid >> 5;
    const int lane = tid & 31;
    const int n  = lane & 15;    // A row (M) for loads; D column (N) for stores
    const int kh = lane >> 4;    // K-half within a 4-wide WMMA step
    float* myrow = &s_cat[wave][lane * CAT_STRIDE];

    // ---- grid-stride loop over 8-tile groups: amortizes the weight staging ----
    for (int tg = blockIdx.x; tg < N_GROUPS; tg += gridDim.x) {
        const int t = tg * WAVES_PB + wave;      // 16-edge tile id (wave-uniform)
        if (t >= N_TILES) continue;              // uniform per wave; EXEC stays full

        const int e  = t * 16 + n;
        const int ie = __builtin_nontemporal_load(iidx + e);   // streamed once
        const int je = __builtin_nontemporal_load(jidx + e);
        const int xi = xin[ie];                                // hot, keep cached
        const int xj = xin[je];

        float rb[6];
        {
            const v2f* rp = (const v2f*)(rbf + (size_t)e * 6);  // 24B stride -> 8B aligned
            v2f r01 = __builtin_nontemporal_load(rp + 0);       // streamed once
            v2f r23 = __builtin_nontemporal_load(rp + 1);
            v2f r45 = __builtin_nontemporal_load(rp + 2);
            rb[0] = r01.x; rb[1] = r01.y; rb[2] = r23.x;
            rb[3] = r23.y; rb[4] = r45.x; rb[5] = r45.y;
        }

        // Accumulators pre-loaded with bias (C[m][n] = b_lin[n], n fixed per lane)
        v8f acc[8];
        #pragma unroll
        for (int nt = 0; nt < 8; ++nt) {
            float bv = s_blin[nt * 16 + n];
            v8f a8;
            #pragma unroll
            for (int r2 = 0; r2 < 8; ++r2) a8[r2] = bv;
            acc[nt] = a8;
        }

        // ---- main GEMM: K=384 in 3 phases of 128 (h_i | h_j | rbf0) ----
        #pragma unroll
        for (int p = 0; p < 3; ++p) {
            if (p < 2) {
                // gather endpoint embedding row; each lane keeps its K-half slice
                const float* row = emb + (size_t)(p == 0 ? xi : xj) * HIDDEN;
                #pragma unroll
                for (int q = 0; q < 32; ++q) {
                    v2f pr = *(const v2f*)(row + 4 * q + 2 * kh);
                    *(v2f*)(myrow + 2 * q) = pr;
                }
            } else {
                // rbf0 = swish(rbf . w0^T + b0), computed directly in A-fragment layout
                #pragma unroll
                for (int q = 0; q < 32; ++q) {
                    int c = 4 * q + 2 * kh;
                    const float* wr0 = &s_w0p[c * 8];   // broadcast reads per half-wave
                    const float* wr1 = &s_w0p[(c + 1) * 8];
                    float z0 = wr0[6] + rb[0]*wr0[0] + rb[1]*wr0[1] + rb[2]*wr0[2]
                                      + rb[3]*wr0[3] + rb[4]*wr0[4] + rb[5]*wr0[5];
                    float z1 = wr1[6] + rb[0]*wr1[0] + rb[1]*wr1[1] + rb[2]*wr1[2]
                                      + rb[3]*wr1[3] + rb[4]*wr1[4] + rb[5]*wr1[5];
                    v2f s;
                    s.x = z0 * __builtin_amdgcn_rcpf(1.0f + __expf(-z0));
                    s.y = z1 * __builtin_amdgcn_rcpf(1.0f + __expf(-z1));
                    *(v2f*)(myrow + 2 * q) = s;
                }
            }
            // 32 x (K=4) WMMA steps over this 128-wide phase, 8 N-tiles of 16 channels
            const float* wk = s_wlin + p * 128 + 2 * kh;
            for (int ks = 0; ks < 32; ++ks) {
                v2f a = *(const v2f*)(myrow + 2 * ks);
                #pragma unroll
                for (int nt = 0; nt < 8; ++nt) {
                    v2f bb = *(const v2f*)(wk + (nt * 16 + n) * WLIN_STRIDE + 4 * ks);
                    acc[nt] = __builtin_amdgcn_wmma_f32_16x16x4_f32(
                        false, a, false, bb, (short)0, acc[nt], false, false);
                }
            }
        }

        // ---- e2 = (rbf . w1^T) * swish(e1-preact); K=6 padded to 8 (2 WMMA steps) ----
        v2f ra0, ra1;
        ra0.x = kh ? rb[2] : rb[0];
        ra0.y = kh ? rb[3] : rb[1];
        ra1.x = kh ? 0.0f  : rb[4];
        ra1.y = kh ? 0.0f  : rb[5];

        #pragma unroll
        for (int nt = 0; nt < 8; ++nt) {
            v8f e1v;
            #pragma unroll
            for (int r2 = 0; r2 < 8; ++r2) {
                float v = acc[nt][r2];
                e1v[r2] = v * __builtin_amdgcn_rcpf(1.0f + __expf(-v));
            }
            const float* wr = &s_w1p[(nt * 16 + n) * 8 + 2 * kh];
            v2f bf0 = *(const v2f*)(wr);
            v2f bf1 = *(const v2f*)(wr + 4);
            v8f r1 = {};
            r1 = __builtin_amdgcn_wmma_f32_16x16x4_f32(false, ra0, false, bf0, (short)0, r1, false, false);
            r1 = __builtin_amdgcn_wmma_f32_16x16x4_f32(false, ra1, false, bf1, (short)0, r1, false, false);
            // D layout: vgpr r2 -> row r2 + 8*kh, col = nt*16 + n
            // Outputs are written once, never re-read -> non-temporal stores
            #pragma unroll
            for (int r2 = 0; r2 < 8; ++r2) {
                size_t addr = (size_t)(t * 16 + 8 * kh + r2) * HIDDEN + nt * 16 + n;
                __builtin_nontemporal_store(e1v[r2], out1 + addr);
                __builtin_nontemporal_store(r1[r2] * e1v[r2], out2 + addr);
            }
        }
    }
}

extern "C" void kernel_launch(void* const* d_in, const int* in_sizes, int n_in,
                              void* d_out, int out_size, void* d_ws, size_t ws_size,
                              hipStream_t stream) {
    (void)in_sizes; (void)n_in; (void)out_size; (void)d_ws; (void)ws_size;
    const int*   x    = (const int*)d_in[0];
    const float* rbf  = (const float*)d_in[1];
    const int*   ii   = (const int*)d_in[2];
    const int*   jj   = (const int*)d_in[3];
    const float* emb  = (const float*)d_in[4];
    const float* w0   = (const float*)d_in[5];
    const float* b0   = (const float*)d_in[6];
    const float* wlin = (const float*)d_in[7];
    const float* blin = (const float*)d_in[8];
    const float* w1   = (const float*)d_in[9];

    float* out1 = (float*)d_out;
    float* out2 = out1 + (size_t)N_EDGES * HIDDEN;

    int blocks = (N_GROUPS < GRID_BLKS) ? N_GROUPS : GRID_BLKS;  // persistent blocks
    edge_mlp_kernel<<<blocks, 256, 0, stream>>>(x, rbf, ii, jj, emb,
                                                w0, b0, wlin, blin, w1,
                                                out1, out2);
}